// ScatterBEV_78262894068390
// MI455X (gfx1250) — compile-verified
//
#include <hip/hip_runtime.h>

// PointPillars BEV scatter for MI455X (gfx1250).
// 0 FLOP/byte -> no WMMA; this is a pure HBM-roofline problem:
//   write 512MB (B,C,H,W) output coalesced+NT, read 61MB features + 8MB winner.
// Ideal @ 23.3 TB/s: ~25us.
// Deterministic duplicate resolution via global_atomic_max_i32 winner map.

namespace {

constexpr int Hh = 512;
constexpr int Ww = 512;
constexpr int HW = Hh * Ww;          // 262144 pixels
constexpr int Bb = 8;
constexpr int Pp = 30000;
constexpr int Cc = 64;

typedef __attribute__((ext_vector_type(4))) float v4f;
typedef __attribute__((ext_vector_type(4))) int   v4i;

// ---------------------------------------------------------------------------
// Phase 1: winner[b][y*W+x] = -1   (8 MB, b128 stores)
// ---------------------------------------------------------------------------
__global__ void winner_init_kernel(int* __restrict__ winner) {
  int i = blockIdx.x * blockDim.x + threadIdx.x;      // over B*HW/4 = 524288
  v4i m1 = { -1, -1, -1, -1 };
  ((v4i*)winner)[i] = m1;
}

// ---------------------------------------------------------------------------
// Phase 2: deterministic duplicate resolution: atomicMax of pillar index.
// One thread per (b,p); coords load is a coalesced b64 per lane.
// ---------------------------------------------------------------------------
__global__ void winner_scatter_kernel(const int2* __restrict__ coords,
                                      int* __restrict__ winner) {
  int i = blockIdx.x * blockDim.x + threadIdx.x;      // over B*P
  if (i >= Bb * Pp) return;
  int b = i / Pp;
  int p = i - b * Pp;
  int2 cc = coords[i];                                 // {gx, gy}
  unsigned gx = (unsigned)cc.x;                        // unsigned compare folds
  unsigned gy = (unsigned)cc.y;                        //  the <0 check
  if (gx < (unsigned)Ww && gy < (unsigned)Hh) {
    atomicMax(winner + (size_t)b * HW + gy * Ww + gx, p);   // global_atomic_max_i32
  }
}

// ---------------------------------------------------------------------------
// Phase 3: gather with fused zero-fill. Thread owns 4 consecutive pixels of
// one batch. It reads its int4 winner once (coalesced b128), prefetches the
// up-to-4 contiguous 256B feature rows (global_prefetch_b8), then loops over
// 16 channel-quads: 4x b128 row loads -> free 4x4 register transpose ->
// 4x coalesced non-temporal b128 output stores. All feature traffic is
// full-cacheline sequential; the 512MB output stream is NT so it does not
// evict the 8MB winner map / feature lines from the 192MB L2.
// ---------------------------------------------------------------------------
__global__ void bev_gather_kernel(const float* __restrict__ feat,
                                  const int* __restrict__ winner,
                                  float* __restrict__ out) {
  int t = blockIdx.x * blockDim.x + threadIdx.x;      // over B*HW/4
  int b    = t / (HW / 4);
  int pix4 = t - b * (HW / 4);

  v4i w = ((const v4i*)winner)[t];                    // coalesced b128 load

  const float* fb = feat + (size_t)b * Pp * Cc;
  // Clamp invalid winners to row 0 (valid address); mask values later.
  const v4f* r0 = (const v4f*)(fb + (size_t)(w.x < 0 ? 0 : w.x) * Cc);
  const v4f* r1 = (const v4f*)(fb + (size_t)(w.y < 0 ? 0 : w.y) * Cc);
  const v4f* r2 = (const v4f*)(fb + (size_t)(w.z < 0 ? 0 : w.z) * Cc);
  const v4f* r3 = (const v4f*)(fb + (size_t)(w.w < 0 ? 0 : w.w) * Cc);

  // gfx1250 prefetch path: warm the 256B feature rows (global_prefetch_b8)
  if (w.x >= 0) __builtin_prefetch((const void*)r0, 0, 3);
  if (w.y >= 0) __builtin_prefetch((const void*)r1, 0, 3);
  if (w.z >= 0) __builtin_prefetch((const void*)r2, 0, 3);
  if (w.w >= 0) __builtin_prefetch((const void*)r3, 0, 3);

  const v4f zv = { 0.0f, 0.0f, 0.0f, 0.0f };
  float* ob = out + (size_t)b * Cc * HW + (size_t)pix4 * 4;

  for (int c4 = 0; c4 < Cc / 4; ++c4) {
    // One b128 load per row covers 4 channels (rows are 256B contiguous).
    v4f a0 = r0[c4];
    v4f a1 = r1[c4];
    v4f a2 = r2[c4];
    v4f a3 = r3[c4];
    a0 = (w.x >= 0) ? a0 : zv;
    a1 = (w.y >= 0) ? a1 : zv;
    a2 = (w.z >= 0) ? a2 : zv;
    a3 = (w.w >= 0) ? a3 : zv;

    // 4x4 register transpose: channel-major vectors across the 4 pixels.
    v4f o0 = { a0.x, a1.x, a2.x, a3.x };
    v4f o1 = { a0.y, a1.y, a2.y, a3.y };
    v4f o2 = { a0.z, a1.z, a2.z, a3.z };
    v4f o3 = { a0.w, a1.w, a2.w, a3.w };

    // Streaming 512MB: non-temporal b128, coalesced across the wave.
    __builtin_nontemporal_store(o0, (v4f*)(ob + (size_t)(4 * c4 + 0) * HW));
    __builtin_nontemporal_store(o1, (v4f*)(ob + (size_t)(4 * c4 + 1) * HW));
    __builtin_nontemporal_store(o2, (v4f*)(ob + (size_t)(4 * c4 + 2) * HW));
    __builtin_nontemporal_store(o3, (v4f*)(ob + (size_t)(4 * c4 + 3) * HW));
  }
}

// ---------------------------------------------------------------------------
// Fallback path if workspace is too small for the winner map (< 8MB):
// zero-fill + direct scatter (plain stores).
// ---------------------------------------------------------------------------
__global__ void zero_out_kernel(float* __restrict__ out) {
  int i = blockIdx.x * blockDim.x + threadIdx.x;      // over B*C*HW/4
  v4f z = { 0.0f, 0.0f, 0.0f, 0.0f };
  __builtin_nontemporal_store(z, (v4f*)out + i);
}

__global__ void direct_scatter_kernel(const float* __restrict__ feat,
                                      const int2* __restrict__ coords,
                                      float* __restrict__ out) {
  int i = blockIdx.x * blockDim.x + threadIdx.x;      // over B*P
  if (i >= Bb * Pp) return;
  int b = i / Pp;
  int2 cc = coords[i];
  unsigned gx = (unsigned)cc.x;
  unsigned gy = (unsigned)cc.y;
  if (gx >= (unsigned)Ww || gy >= (unsigned)Hh) return;
  size_t flat = (size_t)gy * Ww + gx;
  const v4f* fr = (const v4f*)(feat + (size_t)i * Cc);
  float* ob = out + (size_t)b * Cc * HW + flat;
  for (int c4 = 0; c4 < Cc / 4; ++c4) {
    v4f v = fr[c4];                                   // coalesced b128 read
    ob[(size_t)(4 * c4 + 0) * HW] = v.x;
    ob[(size_t)(4 * c4 + 1) * HW] = v.y;
    ob[(size_t)(4 * c4 + 2) * HW] = v.z;
    ob[(size_t)(4 * c4 + 3) * HW] = v.w;
  }
}

} // namespace

extern "C" void kernel_launch(void* const* d_in, const int* in_sizes, int n_in,
                              void* d_out, int out_size, void* d_ws, size_t ws_size,
                              hipStream_t stream) {
  const float* feat   = (const float*)d_in[0];   // (B, P, C) f32
  const int2*  coords = (const int2*)d_in[1];    // (B, P, 2) i32
  float*       out    = (float*)d_out;           // (B, C, H, W) f32

  const int threads = 256;                       // 8 wave32 per block
  const int nPill   = Bb * Pp;                   // 240000
  const size_t winnerBytes = (size_t)Bb * HW * sizeof(int);   // 8 MB

  if (ws_size >= winnerBytes) {
    int* winner = (int*)d_ws;

    // Phase 1: init winner map (B*HW/4 = 524288 int4 stores)
    winner_init_kernel<<<(Bb * HW / 4) / threads, threads, 0, stream>>>(winner);

    // Phase 2: atomic-max scatter of pillar indices
    winner_scatter_kernel<<<(nPill + threads - 1) / threads, threads, 0, stream>>>(
        coords, winner);

    // Phase 3: gather + fused zero-fill, coalesced NT b128 output stream
    bev_gather_kernel<<<(Bb * HW / 4) / threads, threads, 0, stream>>>(
        feat, winner, out);
  } else {
    // Fallback: zero-fill then direct scatter
    zero_out_kernel<<<(Bb * Cc * HW / 4) / threads, threads, 0, stream>>>(out);
    direct_scatter_kernel<<<(nPill + threads - 1) / threads, threads, 0, stream>>>(
        feat, coords, out);
  }
}